// softPaching_74002286510227
// MI455X (gfx1250) — compile-verified
//
#include <hip/hip_runtime.h>
#include <math.h>
#include <stdint.h>

// ---------------------------------------------------------------------------
// out[b, (i*4+j)*32 + c, y, x] = x[b,c,y,x] * p[i][y] * p[j][x]
// p[i][.] = 1-D spectral resample (64 -> 224, forward norm) of the 16-wide
// block indicator starting at 16*i. Exact separability: the Nyquist bin of a
// 16-long run of ones at an even offset vanishes (sum of (-1)^t == 0), so the
// kept 2-D spectrum is Hermitian-separable and irfft2 factors into the product
// of two real 1-D band-limited interpolations (harmonics k = -31..31).
// ---------------------------------------------------------------------------

typedef float v4f __attribute__((ext_vector_type(4)));

#define PI_D 3.14159265358979323846
#define HH   224
#define WW   224
#define BB   4
#define CC   32
#define NUMB 4
#define NTERM 16                       // t-terms per profile sample
#define NPROF (NUMB * HH)              // 896 profile samples
#define PTHREADS (NPROF * NTERM)       // 14336 term-threads

// ---- Kernel 1: profile table via per-term threads + 16-wide LDS reduce ----
// One thread per (i, y, t) Dirichlet term: only 2 f64 sins + 1 f64 div each,
// spread over 56 blocks (many WGPs) instead of 28 heavy waves on one WGP.
__global__ __launch_bounds__(256) void softpatch_profile_kernel(
        float* __restrict__ prof) {
    __shared__ double terms[256];
    const int tid = threadIdx.x;
    const int g   = blockIdx.x * 256 + tid;        // 0..14335
    const int py  = g >> 4;                        // profile sample (i*224+y)
    const int tt  = g & 15;                        // term index within sample
    const int i   = py / HH;                       // block row 0..3
    const int y   = py % HH;                       // output sample 0..223
    const int t   = 16 * i + tt;                   // source sample 0..63

    double d;
    if (2 * y == 7 * t) {
        d = 63.0;                                  // Dirichlet limit at a == 0
    } else {
        double a  = (double)y / 224.0 - (double)t / 64.0;
        double ph = PI_D * a;
        d = sin(63.0 * ph) / sin(ph);              // sum e^{ik*2pi*a}, |k|<=31
    }
    terms[tid] = d;
    __syncthreads();

    if (tt == 0) {                                 // one leader per (i,y)
        double s = 0.0;
        #pragma unroll
        for (int u = 0; u < NTERM; ++u) s += terms[(tid & ~15) + u];
        prof[py] = (float)(s / 64.0);              // forward-norm scaling
    }
}

// ---- Kernel 2: streaming broadcast-multiply scatter -----------------------
// One thread per input float4: 1 NT b128 load, 16 NT b128 stores (411 MB of
// output >> 192 MB L2, so non-temporal is correct). Profile table staged into
// LDS via async global->LDS DMA (CDNA5 ASYNCcnt path), then broadcast-read.
__global__ __launch_bounds__(256) void softpatch_scatter_kernel(
        const float* __restrict__ x,
        const float* __restrict__ prof,
        float* __restrict__ out) {
    __shared__ __align__(16) float lut[NPROF];     // 896 floats = 3.5 KB

    const int tid = threadIdx.x;
    if (tid < 224) {                               // exactly 7 full wave32 waves
        // low 32 bits of a flat shared pointer == raw LDS byte offset
        unsigned lds_off = (unsigned)(uintptr_t)&lut[tid * 4];
        unsigned long long gaddr =
            (unsigned long long)(uintptr_t)(prof + tid * 4);
        asm volatile("global_load_async_to_lds_b128 %0, %1, off"
                     :: "v"(lds_off), "v"(gaddr) : "memory");
        asm volatile("s_wait_asynccnt 0" ::: "memory");
    }
    __syncthreads();

    // n indexes (b, c, y, x4) with x4 = x/4 in [0,56)
    long long n = (long long)blockIdx.x * blockDim.x + tid;
    const int x4 = (int)(n % 56);
    long long r  = n / 56;
    const int y  = (int)(r % HH);
    r /= HH;
    const int c  = (int)(r % CC);
    const int b  = (int)(r / CC);

    const float* ip =
        x + (((size_t)(b * CC + c) * HH + y) * WW + x4 * 4);
    const v4f xin = __builtin_nontemporal_load((const v4f*)ip);

    float ry[4];
    v4f   cx[4];
    #pragma unroll
    for (int i = 0; i < 4; ++i) {
        ry[i] = lut[i * HH + y];                       // scalar row profile
        cx[i] = *(const v4f*)&lut[i * HH + x4 * 4];    // 16B-aligned (896 | 16)
    }

    float* op = out + (((size_t)(b * (NUMB * NUMB * CC) + c) * HH + y) * WW
                       + x4 * 4);
    const size_t kstride = (size_t)CC * HH * WW;       // one mask-channel hop
    #pragma unroll
    for (int i = 0; i < 4; ++i) {
        #pragma unroll
        for (int j = 0; j < 4; ++j) {
            v4f v = xin * (ry[i] * cx[j]);
            __builtin_nontemporal_store(v, (v4f*)op);
            op += kstride;
        }
    }
}

// ---------------------------------------------------------------------------
extern "C" void kernel_launch(void* const* d_in, const int* in_sizes, int n_in,
                              void* d_out, int out_size, void* d_ws, size_t ws_size,
                              hipStream_t stream) {
    const float* x   = (const float*)d_in[0];
    float*       prf = (float*)d_ws;     // 896 floats of scratch
    float*       out = (float*)d_out;    // 4*512*224*224 floats

    softpatch_profile_kernel<<<PTHREADS / 256, 256, 0, stream>>>(prf);

    const int total_quads = BB * CC * HH * (WW / 4);   // 1,605,632
    const int block = 256;
    const int grid  = total_quads / block;             // 6272, exact
    softpatch_scatter_kernel<<<grid, block, 0, stream>>>(x, prf, out);
}